// PostSmoothLayer_80290118632062
// MI455X (gfx1250) — compile-verified
//
#include <hip/hip_runtime.h>
#include <stdint.h>

// ---------------------------------------------------------------------------
// PostSmoothLayer for MI455X (gfx1250, wave32).
// out[b,h,s,d] = x[b,h,s,d] * scale[h*128+d], scale = ones scattered w/ smooth.
// Memory-bound: 268 MB @ 23.3 TB/s ~ 11.5 us. Paths used:
//   - TDM tensor_load_to_lds to stage per-block 256B scale tile into LDS
//   - s_wait_tensorcnt + workgroup barrier
//   - ds_load_b128 scale reads, global_load/store_b128 main stream
//   - v_pk_mul_bf16 packed bf16 VALU (VOP3P)
// ---------------------------------------------------------------------------

typedef __attribute__((ext_vector_type(4))) unsigned int u32x4;
typedef __attribute__((ext_vector_type(8))) int i32x8;
typedef __attribute__((ext_vector_type(4))) int i32x4;

__device__ __forceinline__ uint32_t pk_mul_bf16(uint32_t a, uint32_t b) {
    uint32_t d;
    asm("v_pk_mul_bf16 %0, %1, %2" : "=v"(d) : "v"(a), "v"(b));
    return d;
}

// Build the 4096-entry bf16 channel-scale table in workspace.
// Single block: init -> fence+barrier -> scatter (indices are unique).
__global__ void __launch_bounds__(1024) build_scale_kernel(
    const unsigned short* __restrict__ smooth,   // 2048 bf16 (raw bits)
    const int*            __restrict__ idx,      // 2048 unique in [0,4096)
    unsigned short*       __restrict__ scale)    // 4096 bf16 out (d_ws)
{
    const int t = threadIdx.x;
#pragma unroll
    for (int i = 0; i < 4; ++i)
        scale[t + i * 1024] = 0x3F80u;           // bf16 1.0
    __threadfence();
    __syncthreads();
#pragma unroll
    for (int i = 0; i < 2; ++i) {
        const int k = t + i * 1024;
        scale[idx[k]] = smooth[k];
    }
}

// Main elementwise kernel. Each block: 8192 consecutive bf16 elements
// (64 rows of D=128 within one (b,h) slice -> single h per block).
// 256 threads x 4 chunks x 8 bf16 (b128) per chunk.
__global__ void __launch_bounds__(256) smooth_mul_kernel(
    const unsigned short* __restrict__ x,
    const unsigned short* __restrict__ scale,    // 4096 bf16 table in d_ws
    unsigned short*       __restrict__ out)
{
    __shared__ __align__(16) unsigned short s_scale[128];

    const uint32_t tid        = threadIdx.x;
    const uint64_t block_base = (uint64_t)blockIdx.x * 8192u;
    const uint32_t h          = (blockIdx.x >> 6) & 31u;   // 64 blocks per (b,h) slice

    // --- TDM: DMA this block's 128 scales (256 B) from global into LDS ---
    if ((tid >> 5) == 0) {                        // wave 0 only (EXEC ignored by TDM)
        const uint64_t gaddr = (uint64_t)(uintptr_t)(scale + (size_t)h * 128u);
        const uint32_t lds   = (uint32_t)(uintptr_t)(&s_scale[0]);

        u32x4 g0;
        g0[0] = 1u;                               // count=1, user descriptor
        g0[1] = lds;                              // lds_addr[31:0]
        g0[2] = (uint32_t)gaddr;                  // global_addr[31:0]
        g0[3] = ((uint32_t)(gaddr >> 32) & 0x01FFFFFFu) | (2u << 30); // addr[56:32] | type=2

        i32x8 g1;
        g1[0] = (int)(1u << 16);                  // data_size = 2 bytes
        g1[1] = (int)(128u << 16);                // tensor_dim0 = 128 (bits[79:48] lo)
        g1[2] = (int)(1u   << 16);                // tensor_dim1 = 1   (bits[111:80] lo)
        g1[3] = (int)(128u << 16);                // tile_dim0   = 128 (bits[127:112])
        g1[4] = 1;                                // tile_dim1 = 1, tile_dim2 = 0
        g1[5] = 128;                              // tensor_dim0_stride = 128
        g1[6] = 0;                                // tensor_dim1_stride = 0
        g1[7] = 0;

        const i32x4 gz4 = {0, 0, 0, 0};           // groups 2/3 unused (<=2D tensor)
        const i32x8 gz8 = {0, 0, 0, 0, 0, 0, 0, 0};
        __builtin_amdgcn_tensor_load_to_lds(g0, g1, gz4, gz4, gz8, 0);
        __builtin_amdgcn_s_wait_tensorcnt(0);
    }
    __syncthreads();

    // 8 per-lane scales, read once from LDS (ds_load_b128)
    const uint32_t dbase = (tid * 8u) & 127u;
    const uint4 sv = *(const uint4*)(&s_scale[dbase]);

#pragma unroll
    for (int j = 0; j < 4; ++j) {
        const uint64_t e  = block_base + (uint64_t)j * 2048u + (uint64_t)tid * 8u;
        const uint4    xv = *(const uint4*)(x + e);   // global_load_b128
        uint4 r;
        r.x = pk_mul_bf16(xv.x, sv.x);
        r.y = pk_mul_bf16(xv.y, sv.y);
        r.z = pk_mul_bf16(xv.z, sv.z);
        r.w = pk_mul_bf16(xv.w, sv.w);
        *(uint4*)(out + e) = r;                       // global_store_b128
    }
}

extern "C" void kernel_launch(void* const* d_in, const int* in_sizes, int n_in,
                              void* d_out, int out_size, void* d_ws, size_t ws_size,
                              hipStream_t stream) {
    (void)in_sizes; (void)n_in; (void)out_size; (void)ws_size;
    const unsigned short* x      = (const unsigned short*)d_in[0]; // bf16 [4,32,4096,128]
    const unsigned short* smooth = (const unsigned short*)d_in[1]; // bf16 [2048]
    const int*            idx    = (const int*)d_in[2];            // i32  [2048]
    unsigned short*       out    = (unsigned short*)d_out;         // bf16, same layout
    unsigned short*       scale  = (unsigned short*)d_ws;          // 4096 bf16 (8 KB)

    build_scale_kernel<<<1, 1024, 0, stream>>>(smooth, idx, scale);

    const int total   = 4 * 32 * 4096 * 128;   // 67,108,864 elements
    const int nblocks = total / 8192;          // 8192 blocks
    smooth_mul_kernel<<<nblocks, 256, 0, stream>>>(x, scale, out);
}